// MoE_35837207117928
// MI455X (gfx1250) — compile-verified
//
#include <hip/hip_runtime.h>
#include <hip/hip_bf16.h>

typedef float v2f __attribute__((ext_vector_type(2)));
typedef float v8f __attribute__((ext_vector_type(8)));

#define T_TOK 2048
#define DIM   1024
#define HEXP  512
#define HSH   1024
#define NEXP  16
#define TOPK  4
#define TKC   128            // K-chunk staged in LDS
#define XLDA  (TKC + 4)      // 132 mod 64 == 4 -> conflict-free for 16 lanes
#define HLDA  (HEXP + 4)     // 516 mod 64 == 4

__device__ __forceinline__ v8f wmma_f32(v2f a, v2f b, v8f c) {
  // D = A(16x4 f32) * B(4x16 f32) + C(16x16 f32)
  return __builtin_amdgcn_wmma_f32_16x16x4_f32(false, a, false, b, (short)0, c,
                                               false, false);
}

__device__ __forceinline__ float sigmoidf_(float z) {
  return 1.0f / (1.0f + __expf(-z));
}

// Async global->LDS copy of 16 bytes per lane (gfx1250 ASYNCcnt path).
// Low 32 bits of a generic pointer into LDS == wave-relative LDS offset.
__device__ __forceinline__ void async_copy16(const float* gptr, float* lptr) {
  unsigned lds = (unsigned)(uintptr_t)lptr;
  asm volatile("global_load_async_to_lds_b128 %0, %1, off"
               :: "v"(lds), "v"(gptr) : "memory");
}
__device__ __forceinline__ void async_wait0() {
  asm volatile("s_wait_asynccnt 0x0" ::: "memory");
}

// ---------------- gate: sigmoid(x @ gate_w^T), top-4, normalize -------------
__global__ __launch_bounds__(256) void gate_topk_kernel(
    const float* __restrict__ x, const float* __restrict__ gw,
    float* __restrict__ topw, int* __restrict__ topi) {
  int gid = blockIdx.x * blockDim.x + threadIdx.x;
  int t = gid >> 5;
  int lane = gid & 31;
  if (t >= T_TOK) return;
  const float* xr = x + (size_t)t * DIM;
  float g[NEXP];
  #pragma unroll
  for (int e = 0; e < NEXP; ++e) {
    const float* wr = gw + (size_t)e * DIM;
    float s = 0.f;
    for (int d = lane; d < DIM; d += 32) s += xr[d] * wr[d];
    #pragma unroll
    for (int off = 16; off > 0; off >>= 1) s += __shfl_xor(s, off, 32);
    g[e] = sigmoidf_(s);
  }
  // top-4 selection, identical on all lanes (ties -> lowest index, like jax)
  float w[TOPK]; int id[TOPK];
  unsigned used = 0u;
  float sum = 0.f;
  #pragma unroll
  for (int j = 0; j < TOPK; ++j) {
    float bv = -1.f; int bi = 0;
    #pragma unroll
    for (int e = 0; e < NEXP; ++e) {
      bool ok = !((used >> e) & 1u) && (g[e] > bv);
      bv = ok ? g[e] : bv;
      bi = ok ? e : bi;
    }
    used |= (1u << bi);
    w[j] = bv; id[j] = bi; sum += bv;
  }
  if (lane < TOPK) {
    topw[t * TOPK + lane] = w[lane] / sum;
    topi[t * TOPK + lane] = id[lane];
  }
}

// ---------------- build per-expert token lists ------------------------------
__global__ void scatter_kernel(const int* __restrict__ topi,
                               const float* __restrict__ topw,
                               int* __restrict__ rowmap,
                               float* __restrict__ roww,
                               int* __restrict__ cnt) {
  int i = blockIdx.x * blockDim.x + threadIdx.x;
  if (i >= T_TOK * TOPK) return;
  int e = topi[i];
  int t = i >> 2;
  int pos = atomicAdd(&cnt[e], 1);
  rowmap[e * T_TOK + pos] = t;
  roww[e * T_TOK + pos]  = topw[i];
}

// ---------------- shared expert up: hs = silu(x@ws_in) * (x@vs_in) ----------
__global__ __launch_bounds__(256) void shared_up_kernel(
    const float* __restrict__ x, const float* __restrict__ wsin,
    const float* __restrict__ vsin, float* __restrict__ hs) {
  __shared__ __align__(16) float Xl[16 * XLDA];
  int row0 = blockIdx.x * 16;
  int wave = threadIdx.x >> 5, lane = threadIdx.x & 31;
  int kh = lane >> 4, mn = lane & 15;
  int n0 = blockIdx.y * 128 + wave * 16;
  v8f ah = {}; v8f av = {};
  for (int k0 = 0; k0 < DIM; k0 += TKC) {
    __syncthreads();
    // 16 rows x TKC floats = 512 x 16B pieces, exactly 2 per thread, async
    #pragma unroll
    for (int p = 0; p < 2; ++p) {
      int i = (int)threadIdx.x + p * 256;   // piece index
      int r = i >> 5, c4 = (i & 31) * 4;
      async_copy16(&x[(size_t)(row0 + r) * DIM + k0 + c4], &Xl[r * XLDA + c4]);
    }
    async_wait0();
    __syncthreads();
    for (int kk = 0; kk < TKC; kk += 4) {
      const float* ap = &Xl[mn * XLDA + kk + 2 * kh];
      v2f a; a[0] = ap[0]; a[1] = ap[1];
      int krow = k0 + kk + 2 * kh;
      const float* bh = wsin + (size_t)krow * HSH + n0 + mn;
      const float* bv = vsin + (size_t)krow * HSH + n0 + mn;
      v2f b0; b0[0] = bh[0]; b0[1] = bh[HSH];
      v2f b1; b1[0] = bv[0]; b1[1] = bv[HSH];
      ah = wmma_f32(a, b0, ah);
      av = wmma_f32(a, b1, av);
    }
  }
  #pragma unroll
  for (int r = 0; r < 8; ++r) {
    float hv = ah[r], vv = av[r];
    hs[(size_t)(row0 + r + 8 * kh) * HSH + n0 + mn] = hv * sigmoidf_(hv) * vv;
  }
}

// ---------------- shared expert down: out = hs @ ws_out ---------------------
__global__ __launch_bounds__(256) void shared_down_kernel(
    const float* __restrict__ hs, const float* __restrict__ wsout,
    float* __restrict__ out) {
  __shared__ __align__(16) float Xl[16 * XLDA];
  int row0 = blockIdx.x * 16;
  int wave = threadIdx.x >> 5, lane = threadIdx.x & 31;
  int kh = lane >> 4, mn = lane & 15;
  int n0 = blockIdx.y * 128 + wave * 16;
  v8f acc = {};
  for (int k0 = 0; k0 < HSH; k0 += TKC) {
    __syncthreads();
    #pragma unroll
    for (int p = 0; p < 2; ++p) {
      int i = (int)threadIdx.x + p * 256;
      int r = i >> 5, c4 = (i & 31) * 4;
      async_copy16(&hs[(size_t)(row0 + r) * HSH + k0 + c4], &Xl[r * XLDA + c4]);
    }
    async_wait0();
    __syncthreads();
    for (int kk = 0; kk < TKC; kk += 4) {
      const float* ap = &Xl[mn * XLDA + kk + 2 * kh];
      v2f a; a[0] = ap[0]; a[1] = ap[1];
      int krow = k0 + kk + 2 * kh;
      const float* bp = wsout + (size_t)krow * DIM + n0 + mn;
      v2f b; b[0] = bp[0]; b[1] = bp[DIM];
      acc = wmma_f32(a, b, acc);
    }
  }
  #pragma unroll
  for (int r = 0; r < 8; ++r)
    out[(size_t)(row0 + r + 8 * kh) * DIM + n0 + mn] = acc[r];
}

// ---------------- routed experts: gather / SwiGLU / scatter-add -------------
__global__ __launch_bounds__(256) void routed_kernel(
    const float* __restrict__ x, const float* __restrict__ ewin,
    const float* __restrict__ evin, const float* __restrict__ ewout,
    const int* __restrict__ rowmap, const float* __restrict__ roww,
    const int* __restrict__ cnt, float* __restrict__ out) {
  __shared__ __align__(16) float Xl[16 * XLDA];
  __shared__ __align__(16) float Hl[16 * HLDA];
  __shared__ int   tok[16];
  __shared__ float twt[16];
  int e  = blockIdx.y;
  int t0 = blockIdx.x * 16;
  int c  = cnt[e];
  if (t0 >= c) return;  // block-uniform exit
  if (threadIdx.x < 16) {
    int idx = t0 + (int)threadIdx.x;
    if (idx < c) { tok[threadIdx.x] = rowmap[e * T_TOK + idx];
                   twt[threadIdx.x] = roww[e * T_TOK + idx]; }
    else         { tok[threadIdx.x] = 0; twt[threadIdx.x] = 0.f; }
  }
  __syncthreads();
  int wave = threadIdx.x >> 5, lane = threadIdx.x & 31;
  int kh = lane >> 4, mn = lane & 15;
  const float* Wh = ewin + (size_t)e * DIM * HEXP;
  const float* Wv = evin + (size_t)e * DIM * HEXP;

  // Phase A: hin/vin for this 16-token tile, 4 column tiles per wave (H=512)
  v8f ah[4] = {}; v8f av[4] = {};
  for (int k0 = 0; k0 < DIM; k0 += TKC) {
    __syncthreads();
    #pragma unroll
    for (int p = 0; p < 2; ++p) {
      int i = (int)threadIdx.x + p * 256;
      int r = i >> 5, c4 = (i & 31) * 4;
      async_copy16(&x[(size_t)tok[r] * DIM + k0 + c4], &Xl[r * XLDA + c4]);
    }
    async_wait0();
    __syncthreads();
    for (int kk = 0; kk < TKC; kk += 4) {
      const float* ap = &Xl[mn * XLDA + kk + 2 * kh];
      v2f a; a[0] = ap[0]; a[1] = ap[1];
      int krow = k0 + kk + 2 * kh;
      #pragma unroll
      for (int j = 0; j < 4; ++j) {
        int n0 = (wave * 4 + j) * 16;
        const float* bh = Wh + (size_t)krow * HEXP + n0 + mn;
        const float* bv = Wv + (size_t)krow * HEXP + n0 + mn;
        v2f b0; b0[0] = bh[0]; b0[1] = bh[HEXP];
        v2f b1; b1[0] = bv[0]; b1[1] = bv[HEXP];
        ah[j] = wmma_f32(a, b0, ah[j]);
        av[j] = wmma_f32(a, b1, av[j]);
      }
    }
  }
  __syncthreads();
  #pragma unroll
  for (int j = 0; j < 4; ++j) {
    int n0 = (wave * 4 + j) * 16;
    #pragma unroll
    for (int r = 0; r < 8; ++r) {
      int row = r + 8 * kh;
      float hv = ah[j][r], vv = av[j][r];
      Hl[row * HLDA + n0 + mn] = hv * sigmoidf_(hv) * vv * twt[row];
    }
  }
  __syncthreads();

  // Phase B: out_tile = Hl(16x512) @ Wout_e(512x1024), 8 column tiles per wave
  const float* Wo = ewout + (size_t)e * HEXP * DIM;
  v8f ao[8] = {};
  for (int kk = 0; kk < HEXP; kk += 4) {
    int krow = kk + 2 * kh;
    v2f a; a[0] = Hl[mn * HLDA + krow]; a[1] = Hl[mn * HLDA + krow + 1];
    #pragma unroll
    for (int j = 0; j < 8; ++j) {
      int n0 = wave * 128 + j * 16;
      const float* bp = Wo + (size_t)krow * DIM + n0 + mn;
      v2f b; b[0] = bp[0]; b[1] = bp[DIM];
      ao[j] = wmma_f32(a, b, ao[j]);
    }
  }
  #pragma unroll
  for (int j = 0; j < 8; ++j) {
    int n0 = wave * 128 + j * 16;
    #pragma unroll
    for (int r = 0; r < 8; ++r) {
      int row = r + 8 * kh;
      atomicAdd(&out[(size_t)tok[row] * DIM + n0 + mn], ao[j][r]);
    }
  }
}

extern "C" void kernel_launch(void* const* d_in, const int* in_sizes, int n_in,
                              void* d_out, int out_size, void* d_ws, size_t ws_size,
                              hipStream_t stream) {
  (void)in_sizes; (void)n_in; (void)out_size; (void)ws_size;
  const float* x     = (const float*)d_in[0];
  const float* gw    = (const float*)d_in[1];
  const float* wsin  = (const float*)d_in[2];
  const float* vsin  = (const float*)d_in[3];
  const float* wsout = (const float*)d_in[4];
  const float* ewin  = (const float*)d_in[5];
  const float* evin  = (const float*)d_in[6];
  const float* ewout = (const float*)d_in[7];
  float* out = (float*)d_out;

  // workspace layout (~8.7 MB)
  float* hs     = (float*)d_ws;                       // T*HS
  float* topw   = hs + (size_t)T_TOK * HSH;           // T*K
  int*   topi   = (int*)(topw + T_TOK * TOPK);        // T*K
  int*   rowmap = topi + T_TOK * TOPK;                // E*T
  float* roww   = (float*)(rowmap + NEXP * T_TOK);    // E*T
  int*   cnt    = (int*)(roww + NEXP * T_TOK);        // E

  hipMemsetAsync(cnt, 0, NEXP * sizeof(int), stream);
  gate_topk_kernel<<<T_TOK * 32 / 256, 256, 0, stream>>>(x, gw, topw, topi);
  scatter_kernel<<<(T_TOK * TOPK + 255) / 256, 256, 0, stream>>>(topi, topw,
                                                                 rowmap, roww, cnt);
  shared_up_kernel<<<dim3(T_TOK / 16, HSH / 128), 256, 0, stream>>>(x, wsin, vsin, hs);
  shared_down_kernel<<<dim3(T_TOK / 16, DIM / 128), 256, 0, stream>>>(hs, wsout, out);
  routed_kernel<<<dim3(T_TOK / 16, NEXP), 256, 0, stream>>>(x, ewin, evin, ewout,
                                                            rowmap, roww, cnt, out);
}